// GraphAttentionLayer_63350767616660
// MI455X (gfx1250) — compile-verified
//
#include <hip/hip_runtime.h>
#include <math.h>

// GAT layer: B=4, N=512, F_IN=256, F_OUT=128, H=64  (shapes fixed by reference)
typedef __attribute__((ext_vector_type(2))) float v2f;
typedef __attribute__((ext_vector_type(8))) float v8f;

#define Bb 4
#define Nn 512
#define F_IN 256
#define F_OUT 128
#define Hh 64
#define LRELU_ALPHA 0.01f
#define NEG_INF_F (-1e30f)

// -------------------------------------------------------------------------
// Kernel 1: h[b,n,o] = x[b,n,:] . W[o,:] + bW[o]
// One wave per 16x16 output tile; V_WMMA_F32_16X16X4_F32, K stepped by 4.
// grid (64,4), block 128 (4 waves). 256 tiles per batch (32 M x 8 N).
// -------------------------------------------------------------------------
__global__ void k_gemm_h(const float* __restrict__ x, const float* __restrict__ W,
                         const float* __restrict__ bW, float* __restrict__ h)
{
    const int b    = blockIdx.y;
    const int wid  = blockIdx.x * 4 + (threadIdx.x >> 5);   // 0..255
    const int lane = threadIdx.x & 31;
    const int row  = lane & 15;           // M (for A) / N (for B) within tile
    const int kg   = lane >> 4;           // K-group: 0 or 1

    const int tm = (wid >> 3) << 4;       // 0..496
    const int tn = (wid & 7) << 4;        // 0..112

    const float* xrow = x + ((size_t)b * Nn + tm + row) * F_IN;   // A row
    const float* wrow = W + (size_t)(tn + row) * F_IN;            // B col (W^T)

    v8f c = {};
    for (int k0 = 0; k0 < F_IN; k0 += 4) {
        const int k = k0 + kg * 2;
        v2f a, bf;
        a.x  = xrow[k];  a.y  = xrow[k + 1];
        bf.x = wrow[k];  bf.y = wrow[k + 1];
        c = __builtin_amdgcn_wmma_f32_16x16x4_f32(false, a, false, bf,
                                                  (short)0, c, false, false);
    }
    // C/D layout: vgpr v -> M = tm + v + 8*kg ; lane -> N = tn + row
    float* drow = h + ((size_t)b * Nn + tm + kg * 8) * F_OUT + tn + row;
    const float bias = bW[tn + row];
#pragma unroll
    for (int v = 0; v < 8; ++v)
        drow[(size_t)v * F_OUT] = c[v] + bias;
}

// -------------------------------------------------------------------------
// Kernel 2: hih[m, c] for m in [0,2048): c<64 -> hi = h.A1a^T, c>=64 -> hj = h.A1b^T
// (B,N,F_OUT) is contiguous -> treat as single 2048x128 GEMM, K=128.
// grid (256), block 128. 1024 tiles (128 M x 8 N).
// -------------------------------------------------------------------------
__global__ void k_gemm_hih(const float* __restrict__ h, const float* __restrict__ A1,
                           float* __restrict__ hih)
{
    const int wid  = blockIdx.x * 4 + (threadIdx.x >> 5);   // 0..1023
    const int lane = threadIdx.x & 31;
    const int row  = lane & 15;
    const int kg   = lane >> 4;

    const int tm = (wid >> 3) << 4;       // 0..2032
    const int tn = (wid & 7) << 4;        // 0..112
    const int col = tn + row;             // 0..127

    const float* arow = h + (size_t)(tm + row) * F_OUT;
    // B[k][col] = A1[col&63][(col>>6)*F_OUT + k]   (A1 is H x 2*F_OUT)
    const float* brow = A1 + (size_t)(col & 63) * (2 * F_OUT) + ((col >> 6) * F_OUT);

    v8f c = {};
    for (int k0 = 0; k0 < F_OUT; k0 += 4) {
        const int k = k0 + kg * 2;
        v2f a, bf;
        a.x  = arow[k];  a.y  = arow[k + 1];
        bf.x = brow[k];  bf.y = brow[k + 1];
        c = __builtin_amdgcn_wmma_f32_16x16x4_f32(false, a, false, bf,
                                                  (short)0, c, false, false);
    }
    float* drow = hih + (size_t)(tm + kg * 8) * 128 + col;
#pragma unroll
    for (int v = 0; v < 8; ++v)
        drow[(size_t)v * 128] = c[v];
}

// -------------------------------------------------------------------------
// Kernel 3: logits[b,i,j] = mask(leaky(a2 . relu(hi_i + hj_j + b1) + b2), adj)
// Block per (b,i); hi-row(+b1) and a2 cached in LDS; fused H-reduction.
// grid (512,4), block 256.
// -------------------------------------------------------------------------
__global__ void k_logits(const float* __restrict__ hih, const float* __restrict__ b1,
                         const float* __restrict__ A2, const float* __restrict__ b2,
                         const float* __restrict__ adj, float* __restrict__ P)
{
    __shared__ float shi[Hh];
    __shared__ float sa2[Hh];
    const int b = blockIdx.y, i = blockIdx.x;
    const int t = threadIdx.x;
    if (t < Hh) {
        shi[t] = hih[((size_t)b * Nn + i) * 128 + t] + b1[t];
        sa2[t] = A2[t];
    }
    __syncthreads();
    const float bias2 = b2[0];
    for (int j = t; j < Nn; j += 256) {
        const float* hj = hih + ((size_t)b * Nn + j) * 128 + Hh;
        float acc = 0.f;
#pragma unroll
        for (int hh = 0; hh < Hh; ++hh) {
            const float z = shi[hh] + hj[hh];
            acc = fmaf(sa2[hh], fmaxf(z, 0.f), acc);
        }
        float e = acc + bias2;
        e = (e >= 0.f) ? e : LRELU_ALPHA * e;
        const float a = adj[((size_t)b * Nn + i) * Nn + j];
        P[((size_t)b * Nn + i) * Nn + j] = e * a + (1.f - a) * NEG_INF_F;
    }
}

// -------------------------------------------------------------------------
// Kernel 4: per-batch GLOBAL softmax over all N*N logits (reference reshapes
// to (B, N*N)). Pass1: max; Pass2: exp in place + sum; writes 1/sum.
// grid (4), block 1024.
// -------------------------------------------------------------------------
__global__ void k_softmax(float* __restrict__ P, float* __restrict__ inv_sum)
{
    __shared__ float red[1024];
    const int b = blockIdx.x;
    const int t = threadIdx.x;
    float* base = P + (size_t)b * Nn * Nn;

    float m = -INFINITY;
    for (int idx = t; idx < Nn * Nn; idx += 1024)
        m = fmaxf(m, base[idx]);
    red[t] = m;
    __syncthreads();
    for (int s = 512; s > 0; s >>= 1) {
        if (t < s) red[t] = fmaxf(red[t], red[t + s]);
        __syncthreads();
    }
    const float mx = red[0];
    __syncthreads();

    float sum = 0.f;
    for (int idx = t; idx < Nn * Nn; idx += 1024) {
        const float p = __expf(base[idx] - mx);   // masked entries underflow to 0
        base[idx] = p;
        sum += p;
    }
    red[t] = sum;
    __syncthreads();
    for (int s = 512; s > 0; s >>= 1) {
        if (t < s) red[t] += red[t + s];
        __syncthreads();
    }
    if (t == 0) inv_sum[b] = 1.f / red[0];
}

// -------------------------------------------------------------------------
// Kernel 5: out = elu( (P @ h) * inv_sum )   -- per-batch 512x512 @ 512x128
// WMMA f32 with fused softmax-normalization + ELU epilogue.
// grid (64,4), block 128.
// -------------------------------------------------------------------------
__global__ void k_feature(const float* __restrict__ P, const float* __restrict__ h,
                          const float* __restrict__ inv_sum, float* __restrict__ out)
{
    const int b    = blockIdx.y;
    const int wid  = blockIdx.x * 4 + (threadIdx.x >> 5);
    const int lane = threadIdx.x & 31;
    const int row  = lane & 15;
    const int kg   = lane >> 4;

    const int tm = (wid >> 3) << 4;
    const int tn = (wid & 7) << 4;

    const float* arow = P + ((size_t)b * Nn + tm + row) * Nn;
    const float* bcol = h + (size_t)b * Nn * F_OUT + tn + row;   // B is row-major KxN

    v8f c = {};
    for (int k0 = 0; k0 < Nn; k0 += 4) {
        const int k = k0 + kg * 2;
        v2f a, bf;
        a.x  = arow[k];
        a.y  = arow[k + 1];
        bf.x = bcol[(size_t)k * F_OUT];
        bf.y = bcol[(size_t)(k + 1) * F_OUT];
        c = __builtin_amdgcn_wmma_f32_16x16x4_f32(false, a, false, bf,
                                                  (short)0, c, false, false);
    }
    const float inv = inv_sum[b];
    float* drow = out + ((size_t)b * Nn + tm + kg * 8) * F_OUT + tn + row;
#pragma unroll
    for (int v = 0; v < 8; ++v) {
        const float val = c[v] * inv;
        drow[(size_t)v * F_OUT] = (val > 0.f) ? val : expm1f(val);
    }
}

// -------------------------------------------------------------------------
extern "C" void kernel_launch(void* const* d_in, const int* in_sizes, int n_in,
                              void* d_out, int out_size, void* d_ws, size_t ws_size,
                              hipStream_t stream)
{
    const float* x    = (const float*)d_in[0];   // (4,512,256)
    const float* adj  = (const float*)d_in[1];   // (4,512,512)
    const float* W    = (const float*)d_in[2];   // (128,256)
    const float* bW   = (const float*)d_in[3];   // (128,)
    const float* A1   = (const float*)d_in[4];   // (64,256)
    const float* b1   = (const float*)d_in[5];   // (64,)
    const float* A2   = (const float*)d_in[6];   // (1,64)
    const float* b2   = (const float*)d_in[7];   // (1,)
    float* out = (float*)d_out;                  // (4,512,128)

    // Workspace carve-up (total ~6 MB):
    char* ws = (char*)d_ws;
    float* h     = (float*)(ws);                                   // 4*512*128 = 1 MB
    float* hih   = (float*)(ws + (size_t)Bb * Nn * F_OUT * 4);     // 4*512*128 = 1 MB
    float* P     = (float*)(ws + (size_t)2 * Bb * Nn * F_OUT * 4); // 4*512*512 = 4 MB
    float* stats = (float*)(ws + (size_t)2 * Bb * Nn * F_OUT * 4
                               + (size_t)Bb * Nn * Nn * 4);        // 4 floats

    k_gemm_h  <<<dim3(64, 4),  128,  0, stream>>>(x, W, bW, h);
    k_gemm_hih<<<dim3(256),    128,  0, stream>>>(h, A1, hih);
    k_logits  <<<dim3(Nn, Bb), 256,  0, stream>>>(hih, b1, A2, b2, adj, P);
    k_softmax <<<dim3(Bb),     1024, 0, stream>>>(P, stats);
    k_feature <<<dim3(64, 4),  128,  0, stream>>>(P, h, stats, out);
}